// HausdorffLoss_86698209837164
// MI455X (gfx1250) — compile-verified
//
#include <hip/hip_runtime.h>
#include <hip/hip_bf16.h>

// Problem shape is fixed by the reference: B=32, C=1, H=W=256.
#define B_    32
#define H_    256
#define W_    256
#define NPIX  (B_*H_*W_)      // 2,097,152 elements in the mean
#define NROW  (B_*H_)         // 8192 (b,h) rows
#define CAP   (H_ + W_)       // reference caps 1D distance at H+W = 512
#define BIGI  (1 << 20)

typedef __attribute__((ext_vector_type(2))) float v2f;
typedef __attribute__((ext_vector_type(8))) float v8f;

// ---------------------------------------------------------------------------
// Stage A: per-column exact 1D distance (axis 0) for the 4 masks
//   e=0: pred>0.5   e=1: !(pred>0.5)   e=2: target==1   e=3: target==0
// Two-pass (down then up) scan per column; result stored as capped g^2 (f32).
// One thread per (b, w) column handles all 4 masks to reuse the image loads.
// ---------------------------------------------------------------------------
__global__ void edt_cols_kernel(const float* __restrict__ pred,
                                const float* __restrict__ target,
                                float* __restrict__ g2)
{
    int tid = blockIdx.x * blockDim.x + threadIdx.x;
    if (tid >= B_ * W_) return;
    int b = tid / W_;
    int w = tid - b * W_;

    const size_t img = (size_t)H_ * W_;
    const size_t off = (size_t)b * img + w;
    const float* p = pred   + off;
    const float* t = target + off;
    const size_t BHW = (size_t)B_ * img;
    float* G0 = g2 + 0 * BHW + off;
    float* G1 = g2 + 1 * BHW + off;
    float* G2 = g2 + 2 * BHW + off;
    float* G3 = g2 + 3 * BHW + off;

    // forward pass: distance to nearest feature above (inclusive)
    int d0 = BIGI, d1 = BIGI, d2 = BIGI, d3 = BIGI;
    for (int h = 0; h < H_; ++h) {
        const size_t k = (size_t)h * W_;
        bool m0 = p[k] > 0.5f;
        bool m1 = t[k] == 1.0f;
        d0 = m0  ? 0 : d0 + 1;
        d1 = !m0 ? 0 : d1 + 1;
        d2 = m1  ? 0 : d2 + 1;
        d3 = !m1 ? 0 : d3 + 1;
        G0[k] = (float)d0; G1[k] = (float)d1;
        G2[k] = (float)d2; G3[k] = (float)d3;
    }
    // backward pass: min with distance to nearest feature below, cap, square
    d0 = BIGI; d1 = BIGI; d2 = BIGI; d3 = BIGI;
    for (int h = H_ - 1; h >= 0; --h) {
        const size_t k = (size_t)h * W_;
        bool m0 = p[k] > 0.5f;
        bool m1 = t[k] == 1.0f;
        d0 = m0  ? 0 : d0 + 1;
        d1 = !m0 ? 0 : d1 + 1;
        d2 = m1  ? 0 : d2 + 1;
        d3 = !m1 ? 0 : d3 + 1;
        int g0 = min(min((int)G0[k], d0), CAP);
        int g1 = min(min((int)G1[k], d1), CAP);
        int g3 = min(min((int)G2[k], d2), CAP);
        int g4 = min(min((int)G3[k], d3), CAP);
        G0[k] = (float)(g0 * g0);
        G1[k] = (float)(g1 * g1);
        G2[k] = (float)(g3 * g3);
        G3[k] = (float)(g4 * g4);
    }
}

// ---------------------------------------------------------------------------
// Stage B: per-row fused min-plus (lower envelope, brute force like the ref)
// for all 4 EDTs + loss + deterministic in-block tree reduction.
// One 256-thread (8-wave) block per (b,h) row; g^2 rows staged in LDS (4 KB).
// ---------------------------------------------------------------------------
__global__ void minplus_loss_kernel(const float* __restrict__ g2,
                                    float* __restrict__ rowSums)
{
    __shared__ float s0[W_], s1[W_], s2[W_], s3[W_];
    __shared__ float red[W_];

    const int r = blockIdx.x;             // b*H + h
    const int y = threadIdx.x;            // output column
    const size_t BHW  = (size_t)B_ * H_ * W_;
    const size_t base = (size_t)r * W_;

    s0[y] = g2[0 * BHW + base + y];
    s1[y] = g2[1 * BHW + base + y];
    s2[y] = g2[2 * BHW + base + y];
    s3[y] = g2[3 * BHW + base + y];
    __syncthreads();

    float e0 = 1e30f, e1 = 1e30f, e2 = 1e30f, e3 = 1e30f;
    float dy = (float)y;                  // dy = y - w, decremented per step
#pragma unroll 8
    for (int w = 0; w < W_; ++w) {
        float q = dy * dy;
        e0 = fminf(e0, s0[w] + q);
        e1 = fminf(e1, s1[w] + q);
        e2 = fminf(e2, s2[w] + q);
        e3 = fminf(e3, s3[w] + q);
        dy -= 1.0f;
    }

    float pd   = sqrtf(e0) + sqrtf(e1);   // pred_dist
    float td   = sqrtf(e2) + sqrtf(e3);   // target_dist
    float diff = pd - td;
    float val  = diff * diff * td * td;   // dist_diff * target_dist^ALPHA

    // fixed-order tree reduction (deterministic)
    red[y] = val;
    __syncthreads();
    for (int s = W_ / 2; s > 0; s >>= 1) {
        if (y < s) red[y] += red[y + s];
        __syncthreads();
    }
    if (y == 0) rowSums[r] = red[0];
}

// ---------------------------------------------------------------------------
// Stage C: sum the 8192 row partials with V_WMMA_F32_16X16X4_F32.
// D = A x ones + C sums 64 f32 per WMMA in full f32 precision.
// Single wave32 (EXEC all ones, as WMMA requires); 128 chained WMMAs.
// Column n of the f32 C/D tile lives in lane n (M=0..7) and lane n+16
// (M=8..15), so total = colsum(lane0) + colsum(lane16).
// ---------------------------------------------------------------------------
__global__ void wmma_reduce_kernel(const float* __restrict__ rowSums,
                                   float* __restrict__ out)
{
    const int lane = threadIdx.x;         // 0..31, exactly one wave
    v8f c = {0.f, 0.f, 0.f, 0.f, 0.f, 0.f, 0.f, 0.f};
    v2f ones; ones.x = 1.0f; ones.y = 1.0f;

    const int NT = NROW / 64;             // 128 tiles of 64 f32
    for (int i = 0; i < NT; ++i) {
        const float* p = rowSums + i * 64;
        if (i + 1 < NT)                   // software prefetch of next tile
            __builtin_prefetch(p + 64 + lane, 0, 1);
        v2f a; a.x = p[lane]; a.y = p[lane + 32];
        // (neg_a, A, neg_b, B, c_mod, C, reuse_a, reuse_b)
        c = __builtin_amdgcn_wmma_f32_16x16x4_f32(
                false, a, false, ones, (short)0, c, false, false);
    }

    float s = c[0] + c[1] + c[2] + c[3] + c[4] + c[5] + c[6] + c[7];
    __shared__ float sh[32];
    sh[lane] = s;
    __syncthreads();
    if (lane == 0)
        out[0] = (sh[0] + sh[16]) / (float)NPIX;   // mean
}

// ---------------------------------------------------------------------------
extern "C" void kernel_launch(void* const* d_in, const int* in_sizes, int n_in,
                              void* d_out, int out_size, void* d_ws, size_t ws_size,
                              hipStream_t stream)
{
    (void)in_sizes; (void)n_in; (void)out_size; (void)ws_size;

    const float* pred   = (const float*)d_in[0];
    const float* target = (const float*)d_in[1];
    float* out = (float*)d_out;

    // workspace layout: 4 * [B,H,W] f32 squared 1D distances, then row sums
    float* g2      = (float*)d_ws;
    float* rowSums = g2 + (size_t)4 * NPIX;       // needs ~33.6 MB total

    edt_cols_kernel<<<(B_ * W_ + 255) / 256, 256, 0, stream>>>(pred, target, g2);
    minplus_loss_kernel<<<NROW, W_, 0, stream>>>(g2, rowSums);
    wmma_reduce_kernel<<<1, 32, 0, stream>>>(rowSums, out);
}